// MetricsLoss_65781719106326
// MI455X (gfx1250) — compile-verified
//
#include <hip/hip_runtime.h>
#include <hip/hip_bf16.h>

// MetricsLoss for MI455X (gfx1250, wave32).
//
// Pipeline (all deterministic, no float atomics):
//   K1 dot_kernel      : d[i] = 1 - dot(z_r[i], z_v[i]); per-wave 16 rows,
//                        2 lanes per row (sequential f32x4 streams), final
//                        2-way x 16-row reduction done by ONE
//                        v_wmma_f32_16x16x4_f32 against an all-ones B matrix.
//   K2 label_partials  : per-block partial sums of d*[lab==0], d*[lab==1], counts.
//   K3 group_kernel    : one block per group (512 x 256): mean/std (ddof=1),
//                        z-score MSE, stable bitonic sort by (var_len<<8|idx),
//                        neighbor violation, 256x256 pairwise rank loss.
//   K4 final_kernel    : fixed-order reduction of partials + group losses,
//                        writes [total, l_cdd, l_pcc] ; d already in out+3.

#define NROWS 131072
#define DDIM 512
#define NGROUPS 512
#define GSIZE 256
#define K_MARGIN 0.02f
#define MARGIN 2.0f
#define EPS_F 1e-6f

typedef __attribute__((ext_vector_type(2))) float v2f;
typedef __attribute__((ext_vector_type(8))) float v8f;

__device__ __forceinline__ float block_reduce_sum(float v, float* red) {
    int t = threadIdx.x;
    red[t] = v;
    __syncthreads();
    #pragma unroll
    for (int s = 128; s > 0; s >>= 1) {
        if (t < s) red[t] += red[t + s];
        __syncthreads();
    }
    float r = red[0];
    __syncthreads();
    return r;
}

// ---------------- K1: dot products + WMMA reduction ----------------
// Block = 256 threads = 8 waves; each wave owns 16 consecutive rows.
// Lane l: row = base + (l & 15), half = l >> 4 covers elements [half*256, half*256+256).
// Partial sums sit exactly in the ISA 16x4 f32 A-matrix layout
// (lanes 0-15: K0, lanes 16-31: K2; K1/K3 zeroed). B = ones => D[m,n] = dot(row m).
__global__ void dot_kernel(const float* __restrict__ zr,
                           const float* __restrict__ zv,
                           float* __restrict__ dout) {
    const int wave = threadIdx.x >> 5;
    const int lane = threadIdx.x & 31;
    const int r16  = lane & 15;
    const int half = lane >> 4;
    const int rowbase = blockIdx.x * 128 + wave * 16;
    const int row = rowbase + r16;

    const float4* __restrict__ pr =
        (const float4*)(zr + (size_t)row * DDIM + half * (DDIM / 2));
    const float4* __restrict__ pv =
        (const float4*)(zv + (size_t)row * DDIM + half * (DDIM / 2));

    float s = 0.0f;
    #pragma unroll 8
    for (int i = 0; i < (DDIM / 2) / 4; ++i) {
        float4 a = pr[i];
        float4 b = pv[i];
        s = fmaf(a.x, b.x, s);
        s = fmaf(a.y, b.y, s);
        s = fmaf(a.z, b.z, s);
        s = fmaf(a.w, b.w, s);
    }

    // A[m,:] = [s_m, 0, s_{m+16}, 0]; B = all-ones 4x16; C = 0.
    v2f av; av.x = s;    av.y = 0.0f;
    v2f bv; bv.x = 1.0f; bv.y = 1.0f;
    v8f c = {};
    c = __builtin_amdgcn_wmma_f32_16x16x4_f32(
            /*neg_a=*/false, av, /*neg_b=*/false, bv,
            /*c_mod=*/(short)0, c, /*reuse_a=*/false, /*reuse_b=*/false);

    // C layout: lanes 0-15 hold M=0..7 in c[0..7]; lanes 16-31 hold M=8..15.
    // All N columns are identical, so lane 0 and lane 16 suffice.
    if (lane == 0 || lane == 16) {
        float* o = dout + rowbase + half * 8;
        #pragma unroll
        for (int m = 0; m < 8; ++m) o[m] = 1.0f - c[m];
    }
}

// ---------------- K2: label-masked partial sums ----------------
__global__ void label_partial_kernel(const float* __restrict__ dall,
                                     const int* __restrict__ labels,
                                     float* __restrict__ parts) {
    __shared__ float red[256];
    const int t = threadIdx.x;
    const int b = blockIdx.x;
    float db = 0.0f, dp = 0.0f, nb = 0.0f, np_ = 0.0f;
    for (int i = b * 256 + t; i < NROWS; i += 256 * 256) {
        float dv = dall[i];
        int lab = labels[i];
        if (lab == 0) { db += dv; nb += 1.0f; }
        else          { dp += dv; np_ += 1.0f; }
    }
    db  = block_reduce_sum(db, red);
    dp  = block_reduce_sum(dp, red);
    nb  = block_reduce_sum(nb, red);
    np_ = block_reduce_sum(np_, red);
    if (t == 0) {
        parts[b]       = db;
        parts[256 + b] = dp;
        parts[512 + b] = nb;
        parts[768 + b] = np_;
    }
}

// ---------------- K3: per-group losses ----------------
__global__ void group_kernel(const float* __restrict__ dall,
                             const int* __restrict__ var_lens,
                             float* __restrict__ gloss) {
    __shared__ unsigned keys[GSIZE];
    __shared__ float dd[GSIZE];
    __shared__ float red[GSIZE];

    const int g = blockIdx.x;
    const int t = threadIdx.x;
    const int idx = g * GSIZE + t;

    const float dv = dall[idx];
    const int   vi = var_lens[idx];
    const float vf = (float)vi;

    // ---- stats (order-independent of the later sort) ----
    const float mv = block_reduce_sum(vf, red) * (1.0f / GSIZE);
    const float md = block_reduce_sum(dv, red) * (1.0f / GSIZE);
    const float cv = vf - mv;
    const float cd = dv - md;
    const float sv = sqrtf(block_reduce_sum(cv * cv, red) / (GSIZE - 1));  // ddof=1
    const float sd = sqrtf(block_reduce_sum(cd * cd, red) / (GSIZE - 1));
    const float e  = cv / (sv + EPS_F) - cd / (sd + EPS_F);
    float corr = block_reduce_sum(e * e, red) * (1.0f / GSIZE);
    corr = (sv > 0.0f && sd > 0.0f) ? corr : 0.0f;

    // ---- stable sort by var_len: composite key (v<<8)|idx is unique,
    //      reproducing jnp.argsort's stable tie behavior exactly ----
    keys[t] = ((unsigned)vi << 8) | (unsigned)t;
    dd[t]   = dv;
    __syncthreads();
    for (int k = 2; k <= GSIZE; k <<= 1) {
        for (int j = k >> 1; j > 0; j >>= 1) {
            int ixj = t ^ j;
            if (ixj > t) {
                bool up = ((t & k) == 0);
                unsigned ka = keys[t], kb = keys[ixj];
                if ((ka > kb) == up) {
                    keys[t] = kb; keys[ixj] = ka;
                    float da = dd[t], db2 = dd[ixj];
                    dd[t] = db2; dd[ixj] = da;
                }
            }
            __syncthreads();
        }
    }

    // ---- neighbor monotonicity violation (mean over GSIZE-1) ----
    float nv = 0.0f;
    if (t < GSIZE - 1) nv = fmaxf(dd[t] - dd[t + 1] + K_MARGIN, 0.0f);
    const float neigh = block_reduce_sum(nv, red) * (1.0f / (GSIZE - 1));

    // ---- pairwise rank loss over all ordered pairs with v_s[j] > v_s[t] ----
    const unsigned myv = keys[t] >> 8;
    const float    myd = dd[t];
    float ps = 0.0f;
    float cnt = 0.0f;
    for (int j = 0; j < GSIZE; ++j) {          // LDS broadcast reads
        unsigned vj = keys[j] >> 8;
        if (vj > myv) {
            cnt += 1.0f;
            ps += fmaxf(K_MARGIN - (dd[j] - myd), 0.0f);
        }
    }
    const float psum = block_reduce_sum(ps, red);
    const float csum = block_reduce_sum(cnt, red);
    const float rank = (csum > 0.0f) ? (psum / csum) : 0.0f;

    if (t == 0) gloss[g] = corr + neigh + rank;
}

// ---------------- K4: final combine ----------------
__global__ void final_kernel(const float* __restrict__ parts,
                             const float* __restrict__ gloss,
                             float* __restrict__ out) {
    __shared__ float red[256];
    const int t = threadIdx.x;
    const float db  = block_reduce_sum(parts[t], red);
    const float dp  = block_reduce_sum(parts[256 + t], red);
    const float nb  = block_reduce_sum(parts[512 + t], red);
    const float np_ = block_reduce_sum(parts[768 + t], red);
    const float gl  = block_reduce_sum(gloss[t] + gloss[t + 256], red);
    if (t == 0) {
        const float db_mean = db / fmaxf(nb, 1.0f);
        const float dp_mean = dp / fmaxf(np_, 1.0f);
        const float l_cd  = (nb > 0.0f) ? db_mean : 0.0f;
        const float l_cdd = (nb > 0.0f && np_ > 0.0f)
                                ? fmaxf(MARGIN + db_mean - dp_mean, 0.0f) : 0.0f;
        const float l_pcc = gl * (1.0f / NGROUPS);
        const float LAMBDA_CD = 0.0f, LAMBDA_CDD = 1.0f, LAMBDA_PCC = 1.0f;
        out[0] = LAMBDA_CD * l_cd + LAMBDA_CDD * l_cdd + LAMBDA_PCC * l_pcc;
        out[1] = l_cdd;
        out[2] = l_pcc;
    }
}

extern "C" void kernel_launch(void* const* d_in, const int* in_sizes, int n_in,
                              void* d_out, int out_size, void* d_ws, size_t ws_size,
                              hipStream_t stream) {
    (void)in_sizes; (void)n_in; (void)out_size; (void)ws_size;
    const float* zr       = (const float*)d_in[0];
    const float* zv       = (const float*)d_in[1];
    const int*   labels   = (const int*)d_in[2];
    // d_in[3] = groups (contiguous equal-size, implied by indexing)
    const int*   var_lens = (const int*)d_in[4];

    float* out  = (float*)d_out;
    float* dall = out + 3;                 // d occupies out[3 .. 3+N)

    float* ws    = (float*)d_ws;
    float* parts = ws;                      // 1024 floats
    float* gloss = ws + 1024;               // 512 floats

    dot_kernel<<<NROWS / 128, 256, 0, stream>>>(zr, zv, dall);
    label_partial_kernel<<<256, 256, 0, stream>>>(dall, labels, parts);
    group_kernel<<<NGROUPS, 256, 0, stream>>>(dall, var_lens, gloss);
    final_kernel<<<1, 256, 0, stream>>>(parts, gloss, out);
}